// MixGCN_14697378087207
// MI455X (gfx1250) — compile-verified
//
#include <hip/hip_runtime.h>

// ---------------------------------------------------------------------------
// MixGCN for MI455X (gfx1250):
//  - fp32 WMMA (V_WMMA_F32_16X16X4_F32) GEMMs, LDS-staged A tiles
//    (optionally via GLOBAL_LOAD_ASYNC_TO_LDS_B128 / ASYNCcnt when available)
//  - float4-gather + global_atomic_add_f32 scatter for edge aggregation
// ---------------------------------------------------------------------------

typedef __attribute__((ext_vector_type(2))) float v2f;
typedef __attribute__((ext_vector_type(8))) float v8f;
typedef __attribute__((ext_vector_type(4))) int   v4i;
typedef __attribute__((address_space(1))) v4i     gv4i;   // global
typedef __attribute__((address_space(3))) v4i     lv4i;   // LDS

#if __has_builtin(__builtin_amdgcn_global_load_async_to_lds_b128) && \
    __has_builtin(__builtin_amdgcn_s_wait_asynccnt)
#define USE_ASYNC_STAGE 1
#else
#define USE_ASYNC_STAGE 0
#endif

#define N_NODES 100000
#define N_EDGES 1600000
#define IN_DIM  256
#define HID     128
#define OUTD    64
#define BETA    0.5f
#define CMIX    1.0f

// ------------------------------ utility ------------------------------------

__global__ void fill_f32_kernel(float* __restrict__ p, float v, int n) {
    int i = blockIdx.x * blockDim.x + threadIdx.x;
    if (i < n) p[i] = v;
}

// deg[c] += 1 for every edge target (deg pre-filled with 1.0 for self-loop)
__global__ void degree_kernel(const long long* __restrict__ col,
                              float* __restrict__ deg, int E) {
    int e = blockIdx.x * blockDim.x + threadIdx.x;
    if (e < E) atomicAdd(&deg[(int)col[e]], 1.0f);
}

__global__ void rsqrt_kernel(float* __restrict__ d, int n) {
    int i = blockIdx.x * blockDim.x + threadIdx.x;
    if (i < n) d[i] = rsqrtf(d[i]);   // deg >= 1 always (self-loop)
}

// ------------------------------ WMMA GEMM ----------------------------------
// C[M x NC] = A[M x K] * B[K x NC], fp32.
// Block = 32-row stripe x all NC columns. Waves = NC/16; each wave owns one
// 16-col tile and RT=2 stacked 16-row tiles (B fragment reused for 2 WMMAs).
// A tile staged in LDS (pad 4 floats -> conflict-free ds_load_b64, f4-aligned).
// A-frag (16x4 f32): lanes 0-15 row M=lane K={0,1}; lanes 16-31 K={2,3}.
// B-frag (4x16 f32): lanes hold column N, same K split.
// C/D  (16x16 f32, 8 VGPRs): VGPR r -> M = r + 8*(lane>=16), N = lane%16.
template<int K, int NC, int RT>
__global__ __launch_bounds__(32 * (NC / 16))
void gemm_wmma_kernel(const float* __restrict__ A,
                      const float* __restrict__ B,
                      float* __restrict__ C) {
    constexpr int WAVES   = NC / 16;
    constexpr int THREADS = 32 * WAVES;
    constexpr int ROWS    = 16 * RT;          // rows per block (=32)
    constexpr int PAD     = 4;                // keep float4 align + skew banks
    __shared__ float As[ROWS][K + PAD];

    const int lane    = threadIdx.x & 31;
    const int wave    = threadIdx.x >> 5;     // = column tile index tn
    const int rowBase = blockIdx.x * ROWS;

    // ---- stage A[rowBase .. rowBase+ROWS) x K into LDS ----
    constexpr int VEC_PER_ROW = K / 4;
    constexpr int TOTAL_VEC   = ROWS * VEC_PER_ROW;
#pragma unroll
    for (int v = threadIdx.x; v < TOTAL_VEC; v += THREADS) {
        int r = v / VEC_PER_ROW;
        int c = (v - r * VEC_PER_ROW) * 4;
        const float* gp = A + (size_t)(rowBase + r) * K + c;
#if USE_ASYNC_STAGE
        __builtin_amdgcn_global_load_async_to_lds_b128(
            (gv4i*)gp, (lv4i*)&As[r][c], 0, 0);
#else
        *(float4*)&As[r][c] = *(const float4*)gp;
#endif
    }
#if USE_ASYNC_STAGE
    __builtin_amdgcn_s_wait_asynccnt(0);
#endif
    __syncthreads();

    const int laneLo = lane & 15;
    const int laneHi = lane >> 4;             // 0/1 -> K offset 0/2

    v8f acc[RT];
#pragma unroll
    for (int rt = 0; rt < RT; ++rt) acc[rt] = (v8f){};

    const float* Bcol = B + (size_t)(laneHi * 2) * NC + wave * 16 + laneLo;
#pragma unroll 8
    for (int k = 0; k < K; k += 4) {
        v2f b;
        b.x = Bcol[(size_t)k * NC];
        b.y = Bcol[(size_t)k * NC + NC];
#pragma unroll
        for (int rt = 0; rt < RT; ++rt) {
            v2f a = *(const v2f*)&As[rt * 16 + laneLo][k + laneHi * 2];
            acc[rt] = __builtin_amdgcn_wmma_f32_16x16x4_f32(
                false, a, false, b, (short)0, acc[rt], false, false);
        }
    }

#pragma unroll
    for (int rt = 0; rt < RT; ++rt) {
        float* Crow = C + (size_t)(rowBase + rt * 16 + laneHi * 8) * NC
                        + wave * 16 + laneLo;
#pragma unroll
        for (int r = 0; r < 8; ++r)
            Crow[(size_t)r * NC] = acc[rt][r];
    }
}

// --------------------------- edge scatter-add ------------------------------
// One thread per (edge, 4-feature chunk): AGG[col] += dinv[r]*dinv[c] * H[row].
__global__ void scatter_agg_kernel(const float* __restrict__ H,
                                   const long long* __restrict__ row,
                                   const long long* __restrict__ col,
                                   const float* __restrict__ dinv,
                                   float* __restrict__ AGG,
                                   int E, int chunkShift /* log2(F/4) */, int F) {
    long long tid = (long long)blockIdx.x * blockDim.x + threadIdx.x;
    long long e   = tid >> chunkShift;
    if (e >= E) return;
    int f = (int)(tid & ((1 << chunkShift) - 1)) << 2;

    int r = (int)row[e];
    int c = (int)col[e];
    float nrm = dinv[r] * dinv[c];

    const float4* src = (const float4*)(H + (size_t)r * F + f);
    float* dst = AGG + (size_t)c * F + f;
    __builtin_prefetch(dst, 1, 0);             // global_prefetch_b8
    float4 v = *src;
    atomicAdd(dst + 0, nrm * v.x);
    atomicAdd(dst + 1, nrm * v.y);
    atomicAdd(dst + 2, nrm * v.z);
    atomicAdd(dst + 3, nrm * v.w);
}

// ----------------------------- epilogues -----------------------------------
// z = AGG + dinv^2 * H1 (self-loop) + bias; AGG <- beta*z + (C-beta)*GReLU(z)
__global__ void epilogue1_kernel(float* __restrict__ AGG,
                                 const float* __restrict__ H1,
                                 const float* __restrict__ dinv,
                                 const float* __restrict__ bias,
                                 const float* __restrict__ pga,
                                 const float* __restrict__ pgb,
                                 const float* __restrict__ pgc,
                                 const float* __restrict__ pgd,
                                 int n, int fShift /* log2(F) */) {
    int i = blockIdx.x * blockDim.x + threadIdx.x;
    if (i >= n) return;
    int node = i >> fShift;
    int j    = i & ((1 << fShift) - 1);
    float di = dinv[node];
    float z  = AGG[i] + di * di * H1[i] + bias[j];
    float a = *pga, b = *pgb, c = *pgc, d = *pgd;
    float g = (z < 0.0f) ? a * z : ((z < c) ? b * z : d * z);
    AGG[i] = BETA * z + (CMIX - BETA) * g;
}

// OUT <- CMIX * (OUT + dinv^2 * T2 + bias)
__global__ void epilogue2_kernel(float* __restrict__ OUT,
                                 const float* __restrict__ T2,
                                 const float* __restrict__ dinv,
                                 const float* __restrict__ bias,
                                 int n, int fShift) {
    int i = blockIdx.x * blockDim.x + threadIdx.x;
    if (i >= n) return;
    int node = i >> fShift;
    int j    = i & ((1 << fShift) - 1);
    float di = dinv[node];
    OUT[i] = CMIX * (OUT[i] + di * di * T2[i] + bias[j]);
}

// ------------------------------ launcher -----------------------------------

extern "C" void kernel_launch(void* const* d_in, const int* in_sizes, int n_in,
                              void* d_out, int out_size, void* d_ws, size_t ws_size,
                              hipStream_t stream) {
    const float*     x    = (const float*)d_in[0];
    const long long* ei   = (const long long*)d_in[1];   // int64 [2, E]
    const float*     W1   = (const float*)d_in[2];
    const float*     b1   = (const float*)d_in[3];
    const float*     ga   = (const float*)d_in[4];
    const float*     gb   = (const float*)d_in[5];
    const float*     gc   = (const float*)d_in[6];
    const float*     gd   = (const float*)d_in[7];
    const float*     W2   = (const float*)d_in[8];
    const float*     b2   = (const float*)d_in[9];
    float*           out  = (float*)d_out;

    const long long* row  = ei;
    const long long* col  = ei + N_EDGES;

    // workspace layout (~103 MB): H1 | AGG1 | dinv ; T2 reuses H1's slot
    float* H1   = (float*)d_ws;                       // N*HID
    float* AGG1 = H1 + (size_t)N_NODES * HID;         // N*HID
    float* dinv = AGG1 + (size_t)N_NODES * HID;       // N
    float* T2   = H1;                                 // N*OUTD (reuse after epi1)

    const int T = 256;
    auto blocks = [](long long n, int t) { return (int)((n + t - 1) / t); };

    // 1) degree (init 1.0 for self-loop) -> dinv = rsqrt(deg)
    fill_f32_kernel<<<blocks(N_NODES, T), T, 0, stream>>>(dinv, 1.0f, N_NODES);
    degree_kernel<<<blocks(N_EDGES, T), T, 0, stream>>>(col, dinv, N_EDGES);
    rsqrt_kernel<<<blocks(N_NODES, T), T, 0, stream>>>(dinv, N_NODES);

    // 2) H1 = X @ W1  (WMMA f32)  — 3125 blocks x (8 waves, 32-row stripe)
    gemm_wmma_kernel<IN_DIM, HID, 2><<<N_NODES / 32, 256, 0, stream>>>(x, W1, H1);

    // 3) AGG1 = scatter(norm * H1[row] -> col)
    fill_f32_kernel<<<blocks((long long)N_NODES * HID, T), T, 0, stream>>>(
        AGG1, 0.0f, N_NODES * HID);
    scatter_agg_kernel<<<blocks((long long)N_EDGES * (HID / 4), T), T, 0, stream>>>(
        H1, row, col, dinv, AGG1, N_EDGES, 5 /*log2(128/4)*/, HID);

    // 4) epilogue: self-loop + bias + GReLU mix  (AGG1 becomes H)
    epilogue1_kernel<<<blocks((long long)N_NODES * HID, T), T, 0, stream>>>(
        AGG1, H1, dinv, b1, ga, gb, gc, gd, N_NODES * HID, 7 /*log2(128)*/);

    // 5) T2 = H @ W2  (WMMA f32) — 3125 blocks x (4 waves, 32-row stripe)
    gemm_wmma_kernel<HID, OUTD, 2><<<N_NODES / 32, 128, 0, stream>>>(AGG1, W2, T2);

    // 6) out = scatter(norm * T2[row] -> col) + self-loop + bias
    fill_f32_kernel<<<blocks((long long)N_NODES * OUTD, T), T, 0, stream>>>(
        out, 0.0f, N_NODES * OUTD);
    scatter_agg_kernel<<<blocks((long long)N_EDGES * (OUTD / 4), T), T, 0, stream>>>(
        T2, row, col, dinv, out, N_EDGES, 4 /*log2(64/4)*/, OUTD);
    epilogue2_kernel<<<blocks((long long)N_NODES * OUTD, T), T, 0, stream>>>(
        out, T2, dinv, b2, N_NODES * OUTD, 6 /*log2(64)*/);
}